// End2End_Encoder_63367947485700
// MI455X (gfx1250) — compile-verified
//
#include <hip/hip_runtime.h>
#include <float.h>
#include <math.h>

#define BB 8
#define NN 2048
#define DD 256
#define HH 2048
#define AUXC 6
#define NA (NN - AUXC)        // 2042
#define TOPKC 180
#define RP 192                // padded selected-row count per batch
#define JP 192                // padded kept-col count per batch (186 real)
#define KC 64                 // GEMM K-chunk staged in LDS
#define LDA 68                // LDS row pitch (floats), bank-conflict-free
#define MT 32                 // M rows per block (2 x 16 tiles per wave)

typedef __attribute__((ext_vector_type(2))) float v2f;
typedef __attribute__((ext_vector_type(8))) float v8f;

__device__ __forceinline__ float warp_sum(float v) {
  #pragma unroll
  for (int m = 16; m > 0; m >>= 1) v += __shfl_xor(v, m, 32);
  return v;
}

// --- 1) inverse norm per row: one wave per row ---
__global__ void k_invnorm(const float* __restrict__ xc, float* __restrict__ invn) {
  int wid = threadIdx.x >> 5, lane = threadIdx.x & 31;
  int row = blockIdx.x * 8 + wid;            // [0, B*N)
  const float* xr = xc + (size_t)row * DD;
  float ss = 0.f;
  #pragma unroll
  for (int i = 0; i < 8; ++i) { float v = xr[lane + 32 * i]; ss += v * v; }
  ss = warp_sum(ss);
  if (lane == 0) invn[row] = 1.0f / fmaxf(sqrtf(ss), 1e-12f);
}

// --- 2) mean of normalized rows: one block per batch, thread = d ---
__global__ void k_mean(const float* __restrict__ xc, const float* __restrict__ invn,
                       float* __restrict__ mean) {
  int b = blockIdx.x, d = threadIdx.x;
  const float* xb = xc + (size_t)b * NN * DD;
  const float* ib = invn + b * NN;
  float s = 0.f;
  for (int n = 0; n < NN; ++n) s += xb[(size_t)n * DD + d] * ib[n];
  mean[b * DD + d] = s * (1.0f / (float)NN);
}

// --- 3) scores: one wave per row ---
__global__ void k_score(const float* __restrict__ xc, const float* __restrict__ invn,
                        const float* __restrict__ mean, float* __restrict__ scorebuf,
                        float* __restrict__ keepf, float* __restrict__ out_score) {
  int wid = threadIdx.x >> 5, lane = threadIdx.x & 31;
  int row = blockIdx.x * 8 + wid;
  int b = row / NN, n = row % NN;
  const float* xr = xc + (size_t)row * DD;
  const float* land = xc + ((size_t)b * NN + (NN - 1)) * DD;
  const float* mb = mean + b * DD;
  float d1 = 0.f, d2 = 0.f;
  #pragma unroll
  for (int i = 0; i < 8; ++i) {
    float v = xr[lane + 32 * i];
    d1 += v * land[lane + 32 * i];
    d2 += v * mb[lane + 32 * i];
  }
  d1 = warp_sum(d1); d2 = warp_sum(d2);
  if (lane == 0) {
    float inv_n = invn[row];
    float inv_l = invn[b * NN + (NN - 1)];
    float sdoc = d1 * inv_n * inv_l;           // xn . landmark
    sdoc = sdoc > 0.f ? sdoc : 0.f;
    float ssen = d2 * inv_n;                   // xn . mean(xn)
    ssen = ssen > 0.f ? ssen : 1.0f;
    float sc = 0.9f * sdoc + 0.1f * (1.0f - ssen);
    keepf[row] = (n >= NA) ? 1.0f : 0.0f;
    if (n < NA) {
      scorebuf[b * NN + n] = sc;
      out_score[b * NA + n] = sc;
    }
  }
}

// --- 4) exact top-k (desc value, tie -> lower index), one block per batch ---
__global__ void k_topk(float* __restrict__ scorebuf, float* __restrict__ keepf,
                       int* __restrict__ selIdx, int* __restrict__ jlist,
                       float* __restrict__ out_sel) {
  __shared__ float sv[256];
  __shared__ int   si[256];
  int b = blockIdx.x, t = threadIdx.x;
  float* sb = scorebuf + b * NN;
  for (int r = 0; r < TOPKC; ++r) {
    float bv = -FLT_MAX; int bi = NN;
    for (int n = t; n < NA; n += 256) {
      float v = sb[n];
      if (v > bv) { bv = v; bi = n; }   // n ascending -> first occurrence kept
    }
    sv[t] = bv; si[t] = bi;
    __syncthreads();
    for (int s = 128; s > 0; s >>= 1) {
      if (t < s) {
        float ov = sv[t + s]; int oi = si[t + s];
        if (ov > sv[t] || (ov == sv[t] && oi < si[t])) { sv[t] = ov; si[t] = oi; }
      }
      __syncthreads();
    }
    if (t == 0) {
      int idx = si[0];
      selIdx[b * RP + r] = idx;
      out_sel[b * TOPKC + r] = (float)idx;
      sb[idx] = -FLT_MAX;
      keepf[b * NN + idx] = 1.0f;
    }
    __syncthreads();
  }
  if (t == 0) {
    int c = 0;
    for (int n = 0; n < NN; ++n)
      if (keepf[b * NN + n] > 0.5f) jlist[b * JP + c++] = n;   // ascending, 186 entries
    for (; c < JP; ++c) jlist[b * JP + c] = -1;
  }
  if (t == 1) {
    for (int r = TOPKC; r < RP; ++r) selIdx[b * RP + r] = -1;
  }
}

// --- 5) x_mask output ---
__global__ void k_xmask(const float* __restrict__ keepf, float* __restrict__ out_mask) {
  size_t idx = (size_t)blockIdx.x * 256 + threadIdx.x;   // over B*N*D
  out_mask[idx] = keepf[idx >> 8];
}

// --- 6) gathers: Xg[b][t][:] = x_c[b][jlist[t]][:], Ag[b][r][t] = adj[b][sel[r]][jlist[t]] ---
__global__ void k_gatherX(const float* __restrict__ xc, const int* __restrict__ jlist,
                          float* __restrict__ Xg) {
  int blk = blockIdx.x;
  int b = blk / JP, tI = blk % JP;
  int j = jlist[b * JP + tI];
  float v = 0.f;
  if (j >= 0) v = xc[((size_t)b * NN + j) * DD + threadIdx.x];
  Xg[((size_t)b * JP + tI) * DD + threadIdx.x] = v;
}

__global__ void k_gatherA(const float* __restrict__ adj, const int* __restrict__ selIdx,
                          const int* __restrict__ jlist, float* __restrict__ Ag) {
  int blk = blockIdx.x;
  int b = blk / RP, r = blk % RP;
  int i = selIdx[b * RP + r];
  int tI = threadIdx.x;                      // 192 threads
  int j = jlist[b * JP + tI];
  float v = 0.f;
  if (i >= 0 && j >= 0) v = adj[((size_t)b * NN + i) * NN + j];
  Ag[((size_t)b * RP + r) * JP + tI] = v;
}

// --- 7) f32 WMMA GEMM: C = act(A(MxK) @ B(KxN) + bias) ---
// Block = 8 waves, computes a 32(M) x 256(N) tile; each wave owns 32x32
// (2 M-tiles x 2 N-tiles = 4 accumulators; each B fragment feeds 2 WMMAs).
// A-stripe (32 x KC) staged in LDS (conflict-free pitch); B streamed with
// 32-bit offsets from the uniform base (SADDR + voffset addressing).
template<int ACT, int BIAS>
__global__ void k_gemm_wmma(const float* __restrict__ A, const float* __restrict__ Bm,
                            const float* __restrict__ bias, float* __restrict__ C,
                            int M, int Nn, int K, long sA, long sB, long sC) {
  __shared__ float lA[MT * LDA];             // 8704 B
  int tid = threadIdx.x;
  int wid = tid >> 5, lane = tid & 31;
  int nBase = blockIdx.x * 256 + wid * 32;   // wave's first column (tiles n, n+16)
  int mBase = blockIdx.y * MT;
  const float* Ab = A + (size_t)blockIdx.z * sA;
  const float* Bb = Bm + (size_t)blockIdx.z * sB;
  float* Cb = C + (size_t)blockIdx.z * sC;

  int h  = lane >> 4;                        // K sub-pair selector
  int ml = lane & 15;
  // cooperative A loader mapping: 32 rows x (16 threads * float4), 2 rows/thread
  int lm = tid >> 4;                         // row 0..15 (also loads row lm+16)
  int lc = (tid & 15) * 4;                   // col 0..60
  const float* aRow0 = Ab + (size_t)(mBase + lm) * K + lc;
  const float* aRow1 = Ab + (size_t)(mBase + lm + 16) * K + lc;
  // wave's B column offset (32-bit) for sub-rows (2h, 2h+1)
  unsigned bOff = (unsigned)(2 * h) * (unsigned)Nn + (unsigned)(nBase + ml);

  v8f acc00 = {}, acc01 = {}, acc10 = {}, acc11 = {};
  for (int k0 = 0; k0 < K; k0 += KC) {
    if (k0 + KC < K) {
      __builtin_prefetch(aRow0 + k0 + KC, 0, 1);   // global_prefetch_b8
      __builtin_prefetch(aRow1 + k0 + KC, 0, 1);
    }
    float4 av0 = *(const float4*)(aRow0 + k0);
    float4 av1 = *(const float4*)(aRow1 + k0);
    __syncthreads();                         // previous chunk fully consumed
    *(float4*)&lA[lm * LDA + lc] = av0;      // ds_store_b128
    *(float4*)&lA[(lm + 16) * LDA + lc] = av1;
    __syncthreads();
    #pragma unroll 8
    for (int kk = 0; kk < KC; kk += 4) {
      float2 af0 = *(const float2*)&lA[ml * LDA + kk + 2 * h];        // ds_load_b64
      float2 af1 = *(const float2*)&lA[(16 + ml) * LDA + kk + 2 * h]; // ds_load_b64
      v2f a0;  a0.x = af0.x;  a0.y = af0.y;
      v2f a1;  a1.x = af1.x;  a1.y = af1.y;
      unsigned ro = bOff + (unsigned)(k0 + kk) * (unsigned)Nn;
      v2f b0, b1;
      b0.x = Bb[ro];        b0.y = Bb[ro + Nn];
      b1.x = Bb[ro + 16];   b1.y = Bb[ro + Nn + 16];
      acc00 = __builtin_amdgcn_wmma_f32_16x16x4_f32(false, a0, false, b0,
                                                    (short)0, acc00, false, false);
      acc01 = __builtin_amdgcn_wmma_f32_16x16x4_f32(false, a0, false, b1,
                                                    (short)0, acc01, false, false);
      acc10 = __builtin_amdgcn_wmma_f32_16x16x4_f32(false, a1, false, b0,
                                                    (short)0, acc10, false, false);
      acc11 = __builtin_amdgcn_wmma_f32_16x16x4_f32(false, a1, false, b1,
                                                    (short)0, acc11, false, false);
    }
  }
  unsigned n0 = (unsigned)(nBase + ml);
  float bias0 = 0.f, bias1 = 0.f;
  if (BIAS) { bias0 = bias[n0]; bias1 = bias[n0 + 16]; }
  #pragma unroll
  for (int v = 0; v < 8; ++v) {
    unsigned m0 = (unsigned)(mBase + v + 8 * h);   // C layout: lanes16-31 -> M+8
    unsigned m1 = m0 + 16;
    float c00 = acc00[v], c01 = acc01[v], c10 = acc10[v], c11 = acc11[v];
    if (BIAS) { c00 += bias0; c01 += bias1; c10 += bias0; c11 += bias1; }
    if (ACT) {
      c00 = c00 > 0.f ? c00 : 0.01f * c00;  c01 = c01 > 0.f ? c01 : 0.01f * c01;
      c10 = c10 > 0.f ? c10 : 0.01f * c10;  c11 = c11 > 0.f ? c11 : 0.01f * c11;
    }
    Cb[m0 * (unsigned)Nn + n0]      = c00;
    Cb[m0 * (unsigned)Nn + n0 + 16] = c01;
    Cb[m1 * (unsigned)Nn + n0]      = c10;
    Cb[m1 * (unsigned)Nn + n0 + 16] = c11;
  }
}

// --- 8) background scalar: ((leaky(b1) @ W2) + b2) . Wf + bf ---
__global__ void k_bg(const float* __restrict__ b1, const float* __restrict__ W2,
                     const float* __restrict__ b2, const float* __restrict__ Wf,
                     const float* __restrict__ bfp, float* __restrict__ bg) {
  __shared__ float red[256];
  int d = threadIdx.x;
  float s = b2[d];
  for (int hh = 0; hh < HH; ++hh) {
    float t = b1[hh];
    t = t > 0.f ? t : 0.01f * t;
    s += t * W2[(size_t)hh * DD + d];
  }
  red[d] = s * Wf[d];
  __syncthreads();
  for (int st = 128; st > 0; st >>= 1) {
    if (d < st) red[d] += red[d + st];
    __syncthreads();
  }
  if (d == 0) bg[0] = red[0] + bfp[0];
}

__global__ void k_fillout(const float* __restrict__ bg, float* __restrict__ out) {
  int idx = blockIdx.x * 256 + threadIdx.x;
  if (idx < BB * NA) out[idx] = bg[0];
}

// --- 9) final projection for selected rows, scatter into out ---
__global__ void k_final(const float* __restrict__ A3, const float* __restrict__ Wf,
                        const float* __restrict__ bfp, const int* __restrict__ selIdx,
                        float* __restrict__ out) {
  int wid = threadIdx.x >> 5, lane = threadIdx.x & 31;
  int w = blockIdx.x * 8 + wid;              // [0, B*TOPK)
  int b = w / TOPKC, r = w % TOPKC;
  const float* row = A3 + ((size_t)b * RP + r) * DD;
  float s = 0.f;
  #pragma unroll
  for (int i = 0; i < 8; ++i) s += row[lane + 32 * i] * Wf[lane + 32 * i];
  s = warp_sum(s);
  if (lane == 0) {
    int idx = selIdx[b * RP + r];
    out[b * NA + idx] = s + bfp[0];
  }
}

extern "C" void kernel_launch(void* const* d_in, const int* in_sizes, int n_in,
                              void* d_out, int out_size, void* d_ws, size_t ws_size,
                              hipStream_t stream) {
  (void)in_sizes; (void)n_in; (void)out_size; (void)ws_size;
  const float* xc  = (const float*)d_in[1];
  const float* adj = (const float*)d_in[2];
  const float* Wg  = (const float*)d_in[3];
  const float* W1  = (const float*)d_in[4];
  const float* b1  = (const float*)d_in[5];
  const float* W2  = (const float*)d_in[6];
  const float* b2  = (const float*)d_in[7];
  const float* Wf  = (const float*)d_in[8];
  const float* bfp = (const float*)d_in[9];

  float* out       = (float*)d_out;
  float* out_out   = out;                        // B*NA
  float* out_sel   = out + BB * NA;              // B*TOPK (indices as floats)
  float* out_score = out_sel + BB * TOPKC;       // B*NA
  float* out_mask  = out_score + BB * NA;        // B*N*D

  float* w = (float*)d_ws;
  float* invn     = w; w += BB * NN;
  float* mean     = w; w += BB * DD;
  float* scorebuf = w; w += BB * NN;
  float* keepf    = w; w += BB * NN;
  int*   selIdx   = (int*)w; w += BB * RP;
  int*   jlist    = (int*)w; w += BB * JP;
  float* bg       = w; w += 4;
  float* Ag       = w; w += (size_t)BB * RP * JP;
  float* Xg       = w; w += (size_t)BB * JP * DD;
  float* H0       = w; w += (size_t)BB * RP * DD;
  float* A1       = w; w += (size_t)BB * RP * DD;
  float* A2       = w; w += (size_t)BB * RP * HH;
  float* A3       = w; w += (size_t)BB * RP * DD;

  k_invnorm<<<BB * NN / 8, 256, 0, stream>>>(xc, invn);
  k_mean<<<BB, 256, 0, stream>>>(xc, invn, mean);
  k_score<<<BB * NN / 8, 256, 0, stream>>>(xc, invn, mean, scorebuf, keepf, out_score);
  k_topk<<<BB, 256, 0, stream>>>(scorebuf, keepf, selIdx, jlist, out_sel);
  k_xmask<<<(BB * NN * DD) / 256, 256, 0, stream>>>(keepf, out_mask);
  k_gatherX<<<BB * JP, 256, 0, stream>>>(xc, jlist, Xg);
  k_gatherA<<<BB * RP, 192, 0, stream>>>(adj, selIdx, jlist, Ag);

  // H0 = Ag(192x192) @ Xg(192x256), batched over B
  { dim3 g(DD / 256, RP / MT, BB);
    k_gemm_wmma<0, 0><<<g, 256, 0, stream>>>(Ag, Xg, nullptr, H0, RP, DD, JP,
        (long)RP * JP, (long)JP * DD, (long)RP * DD); }
  // A1 = leaky(H0 @ Wg), flattened M = B*RP
  { dim3 g(DD / 256, (BB * RP) / MT, 1);
    k_gemm_wmma<1, 0><<<g, 256, 0, stream>>>(H0, Wg, nullptr, A1, BB * RP, DD, DD, 0, 0, 0); }
  // A2 = leaky(A1 @ W1 + b1)
  { dim3 g(HH / 256, (BB * RP) / MT, 1);
    k_gemm_wmma<1, 1><<<g, 256, 0, stream>>>(A1, W1, b1, A2, BB * RP, HH, DD, 0, 0, 0); }
  // A3 = A2 @ W2 + b2
  { dim3 g(DD / 256, (BB * RP) / MT, 1);
    k_gemm_wmma<0, 1><<<g, 256, 0, stream>>>(A2, W2, b2, A3, BB * RP, DD, HH, 0, 0, 0); }

  k_bg<<<1, 256, 0, stream>>>(b1, W2, b2, Wf, bfp, bg);
  k_fillout<<<(BB * NA + 255) / 256, 256, 0, stream>>>(bg, out_out);
  k_final<<<(BB * TOPKC) / 8, 256, 0, stream>>>(A3, Wf, bfp, selIdx, out_out);
}